// LSEPLoss_49220325212213
// MI455X (gfx1250) — compile-verified
//
#include <hip/hip_runtime.h>
#include <math.h>

// LSEP loss, factorized:
//   S_b = (sum_{neg} exp(x)) * (sum_{pos} exp(-x));  loss = mean_b log1p(S_b)
// Row sums run on the CDNA5 f32 matrix pipe (V_WMMA_F32_16X16X4_F32) with
// B = ones, so D[m][*] accumulates exact-fp32 row sums of A's K-chunks.
// exp is computed as exp2(x*log2e) on the trans pipe (co-executes with XDL).

typedef float v2f __attribute__((ext_vector_type(2)));
typedef float v8f __attribute__((ext_vector_type(8)));

#define NROWS 512
#define NCOLS 512
#define ROWS_PER_BLOCK 16
#define WAVES_PER_BLOCK 4
#define COLS_PER_WAVE (NCOLS / WAVES_PER_BLOCK)   // 128

#define LOG2E 1.4426950408889634f

__global__ __launch_bounds__(WAVES_PER_BLOCK * 32)
void lsep_rowsum_wmma(const int* __restrict__ y_true,
                      const float* __restrict__ y_pred,
                      float* __restrict__ rowNeg,   // per-row sum exp(x) over label==0
                      float* __restrict__ rowPos)   // per-row sum exp(-x) over label==1
{
    const int tid  = threadIdx.x;
    // Wave index is uniform within a wave32: pin it to an SGPR so the k-loop
    // is scalar-controlled and EXEC stays all-ones around the WMMAs.
    const int wave = __builtin_amdgcn_readfirstlane(tid >> 5);   // 0..3
    const int lane = tid & 31;         // 0..31
    const int r0   = blockIdx.x * ROWS_PER_BLOCK;
    const int row  = r0 + (lane & 15); // A-matrix: M = lane % 16 for both half-waves
    const int kHalf = (lane >> 4) * 2; // lanes 16-31 supply K=2,3 of each K=4 chunk

    const float* predRow = y_pred + (size_t)row * NCOLS;
    const int*   trueRow = y_true + (size_t)row * NCOLS;

    v8f accNeg = {};
    v8f accPos = {};
    const v2f ones = {1.0f, 1.0f};

    const int kBeg = wave * COLS_PER_WAVE;
    const int kEnd = kBeg + COLS_PER_WAVE;

#pragma unroll 4
    for (int k = kBeg; k < kEnd; k += 4) {
        const int kk = k + kHalf;
        const float2 p = *(const float2*)(predRow + kk);
        const int2   t = *(const int2*)(trueRow + kk);

        // exp(x) = exp2(x * log2e): one v_mul + one v_exp_f32, branch-free.
        const float ex0 = __builtin_amdgcn_exp2f(p.x *  LOG2E);
        const float ex1 = __builtin_amdgcn_exp2f(p.y *  LOG2E);
        const float en0 = __builtin_amdgcn_exp2f(p.x * -LOG2E);
        const float en1 = __builtin_amdgcn_exp2f(p.y * -LOG2E);

        v2f aNeg, aPos;                         // selects -> v_cndmask, no branches
        aNeg.x = (t.x == 0) ? ex0 : 0.0f;       // VGPR0 = K even
        aNeg.y = (t.y == 0) ? ex1 : 0.0f;       // VGPR1 = K odd
        aPos.x = (t.x == 1) ? en0 : 0.0f;
        aPos.y = (t.y == 1) ? en1 : 0.0f;

        // D[m][n] += sum_k A[m][k] * 1  -> row sums replicated across N
        accNeg = __builtin_amdgcn_wmma_f32_16x16x4_f32(
            false, aNeg, false, ones, (short)0, accNeg, false, false);
        accPos = __builtin_amdgcn_wmma_f32_16x16x4_f32(
            false, aPos, false, ones, (short)0, accPos, false, false);
    }

    // C/D layout: VGPR m on lanes 0-15 holds M=m; on lanes 16-31 holds M=m+8.
    // Lane 0 (N=0) carries rows 0..7; lane 16 carries rows 8..15.
    __shared__ float sNeg[WAVES_PER_BLOCK][ROWS_PER_BLOCK];
    __shared__ float sPos[WAVES_PER_BLOCK][ROWS_PER_BLOCK];

    if (lane == 0 || lane == 16) {
        const int mBase = (lane >> 4) * 8;
#pragma unroll
        for (int m = 0; m < 8; ++m) {
            sNeg[wave][mBase + m] = accNeg[m];
            sPos[wave][mBase + m] = accPos[m];
        }
    }
    __syncthreads();

    // Combine the 4 column-slice partials (fixed order -> deterministic).
    if (tid < ROWS_PER_BLOCK) {
        float n = sNeg[0][tid] + sNeg[1][tid] + sNeg[2][tid] + sNeg[3][tid];
        float p = sPos[0][tid] + sPos[1][tid] + sPos[2][tid] + sPos[3][tid];
        rowNeg[r0 + tid] = n;
        rowPos[r0 + tid] = p;
    }
}

__global__ __launch_bounds__(NROWS)
void lsep_finalize(const float* __restrict__ rowNeg,
                   const float* __restrict__ rowPos,
                   float* __restrict__ out)
{
    __shared__ float sdata[NROWS];
    const int t = threadIdx.x;
    const float s = rowNeg[t] * rowPos[t];
    sdata[t] = log1pf(s);
    __syncthreads();
#pragma unroll
    for (int stride = NROWS / 2; stride > 0; stride >>= 1) {
        if (t < stride) sdata[t] += sdata[t + stride];
        __syncthreads();
    }
    if (t == 0) out[0] = sdata[0] / (float)NROWS;
}

extern "C" void kernel_launch(void* const* d_in, const int* in_sizes, int n_in,
                              void* d_out, int out_size, void* d_ws, size_t ws_size,
                              hipStream_t stream) {
    (void)in_sizes; (void)n_in; (void)out_size; (void)ws_size;
    const int*   y_true = (const int*)d_in[0];
    const float* y_pred = (const float*)d_in[1];
    float* out    = (float*)d_out;
    float* rowNeg = (float*)d_ws;          // 512 floats
    float* rowPos = rowNeg + NROWS;        // 512 floats

    lsep_rowsum_wmma<<<NROWS / ROWS_PER_BLOCK, WAVES_PER_BLOCK * 32, 0, stream>>>(
        y_true, y_pred, rowNeg, rowPos);
    lsep_finalize<<<1, NROWS, 0, stream>>>(rowNeg, rowPos, out);
}